// Motion3DPerception_20109036880089
// MI455X (gfx1250) — compile-verified
//
#include <hip/hip_runtime.h>
#include <math.h>

// ---------------- constants ----------------
#define NQ      7884
#define N0      6912
#define N1      864
#define N2      108
#define EMBED   128
#define HEADS   4
#define LEVELS  3
#define POINTS  32
#define DH      32
#define NATTN   384    // HEADS * LEVELS * POINTS
#define NOFF    1152   // HEADS * LEVELS * POINTS * 3

// row-tile counts (16 rows / tile)
#define T0 432         // 6912/16
#define T1 54          // 864/16
#define T2 7           // ceil(108/16)
#define TQ 493         // ceil(7884/16)

typedef float v2f __attribute__((ext_vector_type(2)));
typedef float v8f __attribute__((ext_vector_type(8)));

__device__ __forceinline__ v8f vzero8() {
    v8f z;
#pragma unroll
    for (int i = 0; i < 8; ++i) z[i] = 0.0f;
    return z;
}

__device__ __forceinline__ v8f wmma_f32(v2f a, v2f b, v8f c) {
    // V_WMMA_F32_16X16X4_F32 : D = A(16x4,f32) * B(4x16,f32) + C(16x16,f32)
    return __builtin_amdgcn_wmma_f32_16x16x4_f32(
        /*neg_a=*/false, a, /*neg_b=*/false, b,
        /*c_mod=*/(short)0, c, /*reuse_a=*/false, /*reuse_b=*/false);
}

// ---------------- kernel 0: pack weights into K-pair layout ----------------
// packed[(k/2)*N + n] = { W[k][n], W[k+1][n] }  -> one b64 load per B fragment
__global__ __launch_bounds__(256) void pack_pairs_kernel(const float* __restrict__ W,
                                                         v2f* __restrict__ out,
                                                         int K, int N) {
    const int total = (K >> 1) * N;
    const int i = blockIdx.x * blockDim.x + threadIdx.x;
    if (i >= total) return;
    const int kk = i / N;
    const int n  = i % N;
    v2f p;
    p.x = W[(size_t)(2 * kk) * N + n];
    p.y = W[(size_t)(2 * kk + 1) * N + n];
    out[i] = p;
}

// ---------------- kernel 1: fused Q/V level projections ----------------
struct QVParams {
    const float* x1;   // [C, N]  (channel-major)
    const float* x2;   // [C, N]
    const v2f*   wqp;  // packed [C/2, 128]
    const v2f*   wvp;  // packed [C/2, 128]
    const float* bq;   // [128]
    const float* bv;   // [128]
    int N;
    int C;
    int rowbase;       // global row offset of this level inside [NQ, 128]
};

// load next K-step fragments and advance the running pointers
#define QV_LOAD(AQ, AV, BQ, BV)                                              \
    do {                                                                     \
        (AQ).x = a1[0]; (AQ).y = a1[N];                                      \
        (AV).x = a2[0]; (AV).y = a2[N];                                      \
        _Pragma("unroll")                                                    \
        for (int t_ = 0; t_ < 8; ++t_) {                                     \
            (BQ)[t_] = bq[t_ * 16]; (BV)[t_] = bv[t_ * 16];                  \
        }                                                                    \
        a1 += aStep; a2 += aStep; bq += bStep; bv += bStep;                  \
    } while (0)

#define QV_COMPUTE(AQ, AV, BQ, BV)                                           \
    do {                                                                     \
        _Pragma("unroll")                                                    \
        for (int t_ = 0; t_ < 8; ++t_) {                                     \
            accQ[t_] = wmma_f32((AQ), (BQ)[t_], accQ[t_]);                   \
            accV[t_] = wmma_f32((AV), (BV)[t_], accV[t_]);                   \
        }                                                                    \
    } while (0)

__global__ __launch_bounds__(32) void qv_proj_kernel(QVParams p0, QVParams p1, QVParams p2,
                                                     float* __restrict__ q_out,
                                                     float* __restrict__ v_out) {
    int bt = blockIdx.x;
    QVParams p;
    int r0;
    if (bt < T0)            { p = p0; r0 = bt * 16; }
    else if (bt < T0 + T1)  { p = p1; r0 = (bt - T0) * 16; }
    else                    { p = p2; r0 = (bt - T0 - T1) * 16; }

    const int lane = threadIdx.x;
    const int m    = lane & 15;
    const int half = lane >> 4;

    v8f accQ[8], accV[8];
#pragma unroll
    for (int t = 0; t < 8; ++t) { accQ[t] = vzero8(); accV[t] = vzero8(); }

    const int    arow  = min(r0 + m, p.N - 1);  // clamp loads for partial tiles
    const size_t N     = (size_t)p.N;
    const size_t aStep = 4 * N;                 // floats advanced per K-step
    const size_t bStep = 2 * EMBED;             // v2f advanced per K-step

    // running pointers: strength-reduced addressing (adds only, no muls in loop)
    const float* a1 = p.x1 + arow + (size_t)(2 * half) * N;  // {a1[0], a1[N]} = A pair
    const float* a2 = p.x2 + arow + (size_t)(2 * half) * N;
    const v2f*   bq = p.wqp + (size_t)half * EMBED + m;      // + t*16 (immediate)
    const v2f*   bv = p.wvp + (size_t)half * EMBED + m;

    // ---- ping-pong pipelined K loop (no buffer-rotation copies) ----
    // S = C/4 K-steps; C in {64,128,256} -> S even, >= 2
    const int S = p.C >> 2;
    v2f aQ0, aV0, bQ0[8], bV0[8];
    v2f aQ1, aV1, bQ1[8], bV1[8];

    QV_LOAD(aQ0, aV0, bQ0, bV0);                 // step 0 -> buf0
    for (int s = 0; s < S - 2; s += 2) {
        QV_LOAD(aQ1, aV1, bQ1, bV1);             // step s+1 -> buf1
        QV_COMPUTE(aQ0, aV0, bQ0, bV0);          // compute step s
        QV_LOAD(aQ0, aV0, bQ0, bV0);             // step s+2 -> buf0
        QV_COMPUTE(aQ1, aV1, bQ1, bV1);          // compute step s+1
    }
    QV_LOAD(aQ1, aV1, bQ1, bV1);                 // step S-1 -> buf1
    QV_COMPUTE(aQ0, aV0, bQ0, bV0);              // compute step S-2
    QV_COMPUTE(aQ1, aV1, bQ1, bV1);              // compute step S-1

#pragma unroll
    for (int t = 0; t < 8; ++t) {
        const int n     = t * 16 + m;
        const float biq = p.bq[n];
        const float biv = p.bv[n];
#pragma unroll
        for (int r = 0; r < 8; ++r) {
            const int row = r0 + r + 8 * half;
            if (row < p.N) {
                const size_t o = (size_t)(p.rowbase + row) * EMBED + n;
                q_out[o] = accQ[t][r] + biq;
                v_out[o] = accV[t][r] + biv;
            }
        }
    }
}

// ---------------- kernel 2: reference points (with reshape quirk) ----------------
__global__ __launch_bounds__(256) void ref_kernel(const float* __restrict__ f0,
                                                  const float* __restrict__ f1,
                                                  const float* __restrict__ f2,
                                                  float* __restrict__ ref) {
    const int tid = blockIdx.x * blockDim.x + threadIdx.x;
    if (tid >= NQ * 3) return;
    const int n = tid / 3;
    const int j = tid % 3;

    int base, Ri, Ai, Ei;
    const float* flow;
    if (n < N0)            { base = 0;        Ri = 24; Ai = 24; Ei = 12; flow = f0; }
    else if (n < N0 + N1)  { base = N0;       Ri = 12; Ai = 12; Ei = 6;  flow = f1; }
    else                   { base = N0 + N1;  Ri = 6;  Ai = 6;  Ei = 3;  flow = f2; }

    const int nl  = n - base;
    const int N   = Ri * Ai * Ei;
    const int lin = j * N + nl;             // the (3,N) view quirk
    const int c   = lin % 3;                // component used for normalization
    const float dims[3] = {(float)Ri, (float)Ai, (float)Ei};
    const float fv = flow[lin] / dims[c];

    const int e = nl % Ei;
    const int a = (nl / Ei) % Ai;
    const int r = nl / (Ei * Ai);
    float g;
    if (j == 0)      g = ((float)r + 0.5f) / (float)Ri;
    else if (j == 1) g = ((float)a + 0.5f) / (float)Ai;
    else             g = ((float)e + 0.5f) / (float)Ei;

    ref[tid] = g + fv;
}

// ---------------- kernel 3: attn & offset projections ----------------
#define SC_LOAD(A, Bf)                                                       \
    do {                                                                     \
        (A) = ap[0];                                                         \
        _Pragma("unroll")                                                    \
        for (int t_ = 0; t_ < 8; ++t_) (Bf)[t_] = bp[t_ * 16];               \
        ap += 2; bp += bStep;                                                \
    } while (0)

#define SC_COMPUTE(A, Bf)                                                    \
    do {                                                                     \
        _Pragma("unroll")                                                    \
        for (int t_ = 0; t_ < 8; ++t_) acc[t_] = wmma_f32((A), (Bf)[t_], acc[t_]); \
    } while (0)

__global__ __launch_bounds__(32) void score_gemm_kernel(const float* __restrict__ query,
                                                        const v2f* __restrict__ wattp,
                                                        const float* __restrict__ b_attn,
                                                        const v2f* __restrict__ woffp,
                                                        const float* __restrict__ b_off,
                                                        float* __restrict__ attn_out,
                                                        float* __restrict__ off_out) {
    const int bt = blockIdx.x;
    const int ct = bt % 12;
    const int rt = bt / 12;

    const v2f* Wp;
    const float* B;
    float* O;
    int Ncols, colbase;
    if (ct < 3) { Wp = wattp; B = b_attn; O = attn_out; Ncols = NATTN; colbase = ct * 128; }
    else        { Wp = woffp; B = b_off;  O = off_out;  Ncols = NOFF;  colbase = (ct - 3) * 128; }

    const int lane = threadIdx.x;
    const int m    = lane & 15;
    const int half = lane >> 4;
    const int arow = min(rt * 16 + m, NQ - 1);

    // running pointers (adds only in the loop)
    const v2f* ap = (const v2f*)(query + (size_t)arow * EMBED) + half;   // += 2 / step
    const v2f* bp = Wp + (size_t)half * Ncols + colbase + m;             // += 2*Ncols / step
    const size_t bStep = 2 * (size_t)Ncols;

    v8f acc[8];
#pragma unroll
    for (int t = 0; t < 8; ++t) acc[t] = vzero8();

    // S = EMBED/4 = 32 K-steps, ping-pong buffers
    v2f a0, b0[8], a1, b1[8];
    SC_LOAD(a0, b0);
    for (int s = 0; s < (EMBED / 4) - 2; s += 2) {
        SC_LOAD(a1, b1);
        __builtin_prefetch(bp + 4 * bStep, 0, 0);   // global_prefetch_b8, weight stream
        SC_COMPUTE(a0, b0);
        SC_LOAD(a0, b0);
        SC_COMPUTE(a1, b1);
    }
    SC_LOAD(a1, b1);
    SC_COMPUTE(a0, b0);
    SC_COMPUTE(a1, b1);

#pragma unroll
    for (int t = 0; t < 8; ++t) {
        const int n    = colbase + t * 16 + m;
        const float bi = B[n];
#pragma unroll
        for (int r = 0; r < 8; ++r) {
            const int row = rt * 16 + r + 8 * half;
            if (row < NQ) O[(size_t)row * Ncols + n] = acc[t][r] + bi;
        }
    }
}

// ---------------- kernel 4: softmax over 96 logits per (query, head) ----------------
__global__ __launch_bounds__(128) void softmax_kernel(float* __restrict__ attn) {
    const int n    = blockIdx.x;
    const int h    = threadIdx.x >> 5;
    const int lane = threadIdx.x & 31;
    float* base = attn + (size_t)n * NATTN + h * (LEVELS * POINTS);

    float v0 = base[lane], v1 = base[lane + 32], v2 = base[lane + 64];
    float mx = fmaxf(v0, fmaxf(v1, v2));
#pragma unroll
    for (int o = 16; o > 0; o >>= 1) mx = fmaxf(mx, __shfl_xor(mx, o, 32));
    float e0 = __expf(v0 - mx), e1 = __expf(v1 - mx), e2 = __expf(v2 - mx);
    float s = e0 + e1 + e2;
#pragma unroll
    for (int o = 16; o > 0; o >>= 1) s += __shfl_xor(s, o, 32);
    const float inv = 1.0f / s;
    base[lane]      = e0 * inv;
    base[lane + 32] = e1 * inv;
    base[lane + 64] = e2 * inv;
}

// ---------------- kernel 5: trilinear sampling + attention aggregation ----------------
__global__ __launch_bounds__(128) void sample_kernel(const float* __restrict__ value,
                                                     const float* __restrict__ ref,
                                                     const float* __restrict__ off,
                                                     const float* __restrict__ attn,
                                                     float* __restrict__ samp) {
    const int n = blockIdx.x;
    const int h = threadIdx.x >> 5;   // head
    const int d = threadIdx.x & 31;   // channel within head (DH == 32 == wave width)

    const float r0 = ref[n * 3 + 0];
    const float r1 = ref[n * 3 + 1];
    const float r2 = ref[n * 3 + 2];
    const float* offn  = off  + (size_t)n * NOFF;
    const float* attnn = attn + (size_t)n * NATTN + h * (LEVELS * POINTS);

    const int RA[3] = {24, 12, 6};
    const int AA[3] = {24, 12, 6};
    const int EA[3] = {12, 6, 3};
    const int ST[3] = {0, N0, N0 + N1};

    float acc = 0.0f;
#pragma unroll
    for (int l = 0; l < LEVELS; ++l) {
        const int Ri = RA[l], Ai = AA[l], Ei = EA[l], st = ST[l];
        const float Rf = (float)Ri, Af = (float)Ai, Ef = (float)Ei;
        for (int p = 0; p < POINTS; ++p) {
            const int oc = ((h * LEVELS + l) * POINTS + p) * 3;
            const float aw = attnn[l * POINTS + p];
            // loc = ref + off/shape ; pixel coord = loc*dim - 0.5
            const float x = r0 * Rf + offn[oc + 0] - 0.5f;
            const float y = r1 * Af + offn[oc + 1] - 0.5f;
            const float z = r2 * Ef + offn[oc + 2] - 0.5f;
            const float xf = floorf(x), yf = floorf(y), zf = floorf(z);
            const float fx = x - xf, fy = y - yf, fz = z - zf;
            const int x0 = (int)xf, y0 = (int)yf, z0 = (int)zf;
#pragma unroll
            for (int dx = 0; dx < 2; ++dx)
#pragma unroll
            for (int dy = 0; dy < 2; ++dy)
#pragma unroll
            for (int dz = 0; dz < 2; ++dz) {
                const int xi = x0 + dx, yi = y0 + dy, zi = z0 + dz;
                const bool valid = (xi >= 0) & (xi < Ri) & (yi >= 0) & (yi < Ai) &
                                   (zi >= 0) & (zi < Ei);
                const int xc = min(max(xi, 0), Ri - 1);
                const int yc = min(max(yi, 0), Ai - 1);
                const int zc = min(max(zi, 0), Ei - 1);
                const float w = (dx ? fx : 1.0f - fx) * (dy ? fy : 1.0f - fy) *
                                (dz ? fz : 1.0f - fz);
                const int idx = (xc * Ai + yc) * Ei + zc;
                const float g = value[(size_t)(st + idx) * EMBED + h * DH + d];
                acc += aw * w * (valid ? g : 0.0f);
            }
        }
    }
    samp[(size_t)n * EMBED + h * DH + d] = acc;
}

// ---------------- kernel 6: output projection + bias + residual ----------------
__global__ __launch_bounds__(32) void out_gemm_kernel(const float* __restrict__ samp,
                                                      const v2f* __restrict__ woutp,
                                                      const float* __restrict__ b_out,
                                                      const float* __restrict__ query,
                                                      float* __restrict__ out) {
    const int rt   = blockIdx.x;
    const int lane = threadIdx.x;
    const int m    = lane & 15;
    const int half = lane >> 4;
    const int arow = min(rt * 16 + m, NQ - 1);

    const v2f* ap = (const v2f*)(samp + (size_t)arow * EMBED) + half;
    const v2f* bp = woutp + (size_t)half * EMBED + m;
    const size_t bStep = 2 * (size_t)EMBED;

    v8f acc[8];
#pragma unroll
    for (int t = 0; t < 8; ++t) acc[t] = vzero8();

    v2f a0, b0[8], a1, b1[8];
    SC_LOAD(a0, b0);
    for (int s = 0; s < (EMBED / 4) - 2; s += 2) {
        SC_LOAD(a1, b1);
        SC_COMPUTE(a0, b0);
        SC_LOAD(a0, b0);
        SC_COMPUTE(a1, b1);
    }
    SC_LOAD(a1, b1);
    SC_COMPUTE(a0, b0);
    SC_COMPUTE(a1, b1);

#pragma unroll
    for (int t = 0; t < 8; ++t) {
        const int n    = t * 16 + m;
        const float bi = b_out[n];
#pragma unroll
        for (int r = 0; r < 8; ++r) {
            const int row = rt * 16 + r + 8 * half;
            if (row < NQ) {
                const size_t o = (size_t)row * EMBED + n;
                out[o] = acc[t][r] + bi + query[o];
            }
        }
    }
}

// ---------------- launcher ----------------
static inline void launch_pack(const float* W, v2f* out, int K, int N, hipStream_t s) {
    const int total = (K / 2) * N;
    pack_pairs_kernel<<<(total + 255) / 256, 256, 0, s>>>(W, out, K, N);
}

extern "C" void kernel_launch(void* const* d_in, const int* in_sizes, int n_in,
                              void* d_out, int out_size, void* d_ws, size_t ws_size,
                              hipStream_t stream) {
    (void)in_sizes; (void)n_in; (void)out_size; (void)ws_size;

    // setup_inputs() dict order
    const float* x1_0 = (const float*)d_in[0];
    const float* x2_0 = (const float*)d_in[1];
    const float* fl_0 = (const float*)d_in[2];
    const float* wq_0 = (const float*)d_in[3];
    const float* bq_0 = (const float*)d_in[4];
    const float* wv_0 = (const float*)d_in[5];
    const float* bv_0 = (const float*)d_in[6];
    const float* x1_1 = (const float*)d_in[7];
    const float* x2_1 = (const float*)d_in[8];
    const float* fl_1 = (const float*)d_in[9];
    const float* wq_1 = (const float*)d_in[10];
    const float* bq_1 = (const float*)d_in[11];
    const float* wv_1 = (const float*)d_in[12];
    const float* bv_1 = (const float*)d_in[13];
    const float* x1_2 = (const float*)d_in[14];
    const float* x2_2 = (const float*)d_in[15];
    const float* fl_2 = (const float*)d_in[16];
    const float* wq_2 = (const float*)d_in[17];
    const float* bq_2 = (const float*)d_in[18];
    const float* wv_2 = (const float*)d_in[19];
    const float* bv_2 = (const float*)d_in[20];
    const float* w_attn = (const float*)d_in[21];
    const float* b_attn = (const float*)d_in[22];
    const float* w_off  = (const float*)d_in[23];
    const float* b_off  = (const float*)d_in[24];
    const float* w_out  = (const float*)d_in[25];
    const float* b_out  = (const float*)d_in[26];

    // workspace carve-up (floats; ref chunk is odd -> later v2f buffers start at odd float offsets)
    float* ws    = (float*)d_ws;
    float* q     = ws;                    ws += (size_t)NQ * EMBED;
    float* v     = ws;                    ws += (size_t)NQ * EMBED;
    float* ref   = ws;                    ws += (size_t)NQ * 3 + 1;   // +1 pad float makes size odd
    float* attn  = ws;                    ws += (size_t)NQ * NATTN;
    float* off   = ws;                    ws += (size_t)NQ * NOFF;
    float* samp  = ws;                    ws += (size_t)NQ * EMBED;
    v2f* pq0  = (v2f*)ws;  ws += (size_t)64  * EMBED;
    v2f* pv0  = (v2f*)ws;  ws += (size_t)64  * EMBED;
    v2f* pq1  = (v2f*)ws;  ws += (size_t)128 * EMBED;
    v2f* pv1  = (v2f*)ws;  ws += (size_t)128 * EMBED;
    v2f* pq2  = (v2f*)ws;  ws += (size_t)256 * EMBED;
    v2f* pv2  = (v2f*)ws;  ws += (size_t)256 * EMBED;
    v2f* patt = (v2f*)ws;  ws += (size_t)128 * NATTN;
    v2f* poff = (v2f*)ws;  ws += (size_t)128 * NOFF;
    v2f* pout = (v2f*)ws;  ws += (size_t)128 * EMBED;
    float* out = (float*)d_out;

    // one-shot weight repack into K-pair layout
    launch_pack(wq_0, pq0, 64,  EMBED, stream);
    launch_pack(wv_0, pv0, 64,  EMBED, stream);
    launch_pack(wq_1, pq1, 128, EMBED, stream);
    launch_pack(wv_1, pv1, 128, EMBED, stream);
    launch_pack(wq_2, pq2, 256, EMBED, stream);
    launch_pack(wv_2, pv2, 256, EMBED, stream);
    launch_pack(w_attn, patt, 128, NATTN, stream);
    launch_pack(w_off,  poff, 128, NOFF,  stream);
    launch_pack(w_out,  pout, 128, EMBED, stream);

    QVParams p0 = {x1_0, x2_0, pq0, pv0, bq_0, bv_0, N0, 64,  0};
    QVParams p1 = {x1_1, x2_1, pq1, pv1, bq_1, bv_1, N1, 128, N0};
    QVParams p2 = {x1_2, x2_2, pq2, pv2, bq_2, bv_2, N2, 256, N0 + N1};

    qv_proj_kernel<<<T0 + T1 + T2, 32, 0, stream>>>(p0, p1, p2, q, v);
    ref_kernel<<<(NQ * 3 + 255) / 256, 256, 0, stream>>>(fl_0, fl_1, fl_2, ref);
    score_gemm_kernel<<<TQ * 12, 32, 0, stream>>>(q, patt, b_attn, poff, b_off, attn, off);
    softmax_kernel<<<NQ, 128, 0, stream>>>(attn);
    sample_kernel<<<NQ, 128, 0, stream>>>(v, ref, off, attn, samp);
    out_gemm_kernel<<<TQ, 32, 0, stream>>>(samp, pout, b_out, q, out);
}